// VoxelFeatureEncodingLayer_45784351375624
// MI455X (gfx1250) — compile-verified
//
#include <hip/hip_runtime.h>

// -------- CDNA5 WMMA types --------
typedef __attribute__((ext_vector_type(2))) float v2f;
typedef __attribute__((ext_vector_type(8))) float v8f;

#define V_NUM     65536
#define P_NUM     32
#define C_IN      10
#define C_OUT     64
#define K_PAD     12            // 10 features + 1.0 (bias lane) + 0 pad -> 3 k-steps of 4
#define CHUNK_PTS 128           // 4 voxels per chunk, 8 waves x 16 points
#define N_CHUNKS  (V_NUM * P_NUM / CHUNK_PTS)   // 16384
#define GRID1     1024
#define NTHREADS  256
#define BN_EPS    1e-5f

__device__ __forceinline__ v8f wmma_k4(v2f a, v2f b, v8f c) {
  // D = A(16x4,f32) * B(4x16,f32) + C(16x16,f32)  -> v_wmma_f32_16x16x4_f32
  return __builtin_amdgcn_wmma_f32_16x16x4_f32(false, a, false, b, (short)0, c,
                                               false, false);
}

// Stage padded weights into LDS: wlds[n][k], k<10 -> W[n][k], k==10 -> b[n], k==11 -> 0
__device__ __forceinline__ void stage_weights(float* wlds, const float* W,
                                              const float* b, int t) {
  for (int idx = t; idx < C_OUT * K_PAD; idx += NTHREADS) {
    int n = idx / K_PAD, k = idx - n * K_PAD;
    float w = 0.0f;
    if (k < C_IN) w = W[n * C_IN + k];
    else if (k == C_IN) w = b[n];
    wlds[idx] = w;
  }
}

// Load B operands (4 N-tiles x 3 k-steps) into registers from LDS.
// B layout (4x16, f32): lane = 16*khalf + n, VGPR j holds K = 2*khalf + j.
__device__ __forceinline__ void load_breg(const float* wlds, v2f breg[4][3],
                                          int m, int khalf) {
#pragma unroll
  for (int tl = 0; tl < 4; ++tl)
#pragma unroll
    for (int ks = 0; ks < 3; ++ks) {
      const float* p = &wlds[(tl * 16 + m) * K_PAD + ks * 4 + 2 * khalf];
      v2f bb;
      bb.x = p[0];
      bb.y = p[1];
      breg[tl][ks] = bb;
    }
}

// Stage one 128-point chunk into LDS, padded to K_PAD, invalid slots zeroed
// (including the bias-1.0 lane, so invalid slots produce x == 0 exactly).
__device__ __forceinline__ void stage_features(float* flds, const float* vf,
                                               const int* vnp, int g0, int t) {
  for (int idx = t; idx < CHUNK_PTS * K_PAD; idx += NTHREADS) {
    int pt = idx / K_PAD, k = idx - pt * K_PAD;
    int g = g0 + pt;
    int vx = g >> 5;
    int pp = g & 31;
    bool valid = pp < vnp[vx];
    float val = 0.0f;
    if (k < C_IN) val = vf[g * C_IN + k];
    else if (k == C_IN) val = 1.0f;
    flds[idx] = valid ? val : 0.0f;
  }
}

// ---------------- Pass 1: x = feat*W^T + b (WMMA) ; masked  sum / sumsq ------
__global__ __launch_bounds__(NTHREADS) void vfe_stats_kernel(
    const float* __restrict__ vf, const int* __restrict__ vnp,
    const float* __restrict__ W, const float* __restrict__ b,
    float* __restrict__ psum, float* __restrict__ psq) {
  __shared__ __align__(16) float wlds[C_OUT * K_PAD];
  __shared__ __align__(16) float flds[CHUNK_PTS * K_PAD];
  __shared__ float redS[8][C_OUT];
  __shared__ float redQ[8][C_OUT];

  const int t = threadIdx.x;
  const int wave = t >> 5;
  const int lane = t & 31;
  const int m = lane & 15;
  const int khalf = lane >> 4;

  stage_weights(wlds, W, b, t);
  __syncthreads();

  v2f breg[4][3];
  load_breg(wlds, breg, m, khalf);

  float s_acc[4] = {0.f, 0.f, 0.f, 0.f};
  float q_acc[4] = {0.f, 0.f, 0.f, 0.f};

  for (int c = blockIdx.x; c < N_CHUNKS; c += gridDim.x) {
    const int g0 = c * CHUNK_PTS;
    stage_features(flds, vf, vnp, g0, t);
    __syncthreads();

    v8f z = {};
    v8f acc[4] = {z, z, z, z};
    const float* arow = &flds[(wave * 16 + m) * K_PAD];
#pragma unroll
    for (int ks = 0; ks < 3; ++ks) {
      // A layout (16x4, f32): lane = 16*khalf + m, VGPR j holds K = 2*khalf + j
      v2f a;
      a.x = arow[ks * 4 + 2 * khalf];
      a.y = arow[ks * 4 + 2 * khalf + 1];
#pragma unroll
      for (int tl = 0; tl < 4; ++tl) acc[tl] = wmma_k4(a, breg[tl][ks], acc[tl]);
    }

    // Invalid slots give x == 0 exactly -> contribute nothing to sum / sumsq.
#pragma unroll
    for (int tl = 0; tl < 4; ++tl) {
      float s = 0.f, q = 0.f;
#pragma unroll
      for (int r = 0; r < 8; ++r) {
        float x = acc[tl][r];
        s += x;
        q += x * x;
      }
      s_acc[tl] += s;
      q_acc[tl] += q;
    }
    __syncthreads();  // before next chunk overwrites flds
  }

  // Lanes L and L+16 hold the same channel (different M halves): fold halves.
#pragma unroll
  for (int tl = 0; tl < 4; ++tl) {
    s_acc[tl] += __shfl_xor(s_acc[tl], 16);
    q_acc[tl] += __shfl_xor(q_acc[tl], 16);
  }
  if (lane < 16) {
#pragma unroll
    for (int tl = 0; tl < 4; ++tl) {
      redS[wave][tl * 16 + lane] = s_acc[tl];
      redQ[wave][tl * 16 + lane] = q_acc[tl];
    }
  }
  __syncthreads();
  if (t < C_OUT) {
    float s = 0.f, q = 0.f;
#pragma unroll
    for (int w = 0; w < 8; ++w) {
      s += redS[w][t];
      q += redQ[w][t];
    }
    psum[blockIdx.x * C_OUT + t] = s;
    psq[blockIdx.x * C_OUT + t] = q;
  }
}

// ---------------- Pass 2: reduce partials -> scale/shift per channel --------
__global__ __launch_bounds__(NTHREADS) void vfe_finalize_kernel(
    const int* __restrict__ vnp, const float* __restrict__ gamma,
    const float* __restrict__ beta, const float* __restrict__ psum,
    const float* __restrict__ psq, float* __restrict__ sshift) {
  __shared__ float rs[NTHREADS];
  __shared__ float rq[NTHREADS];
  __shared__ int rc[NTHREADS];
  const int ch = blockIdx.x;
  const int t = threadIdx.x;

  float s = 0.f, q = 0.f;
  for (int i = t; i < GRID1; i += NTHREADS) {
    s += psum[i * C_OUT + ch];
    q += psq[i * C_OUT + ch];
  }
  int cnt = 0;
  for (int j = t; j < V_NUM; j += NTHREADS) {
    int np = vnp[j];
    cnt += (np < P_NUM) ? np : P_NUM;
  }
  rs[t] = s;
  rq[t] = q;
  rc[t] = cnt;
  __syncthreads();
  for (int off = NTHREADS / 2; off > 0; off >>= 1) {
    if (t < off) {
      rs[t] += rs[t + off];
      rq[t] += rq[t + off];
      rc[t] += rc[t + off];
    }
    __syncthreads();
  }
  if (t == 0) {
    float n = fmaxf((float)rc[0], 1.0f);
    float mean = rs[0] / n;
    float var = fmaxf(rq[0] / n - mean * mean, 0.0f);
    float sc = gamma[ch] * rsqrtf(var + BN_EPS);
    sshift[ch] = sc;
    sshift[C_OUT + ch] = beta[ch] - mean * sc;
  }
}

// ---------------- Pass 3: recompute x (WMMA), normalize+relu+mask, voxel mean
__global__ __launch_bounds__(NTHREADS) void vfe_output_kernel(
    const float* __restrict__ vf, const int* __restrict__ vnp,
    const float* __restrict__ W, const float* __restrict__ b,
    const float* __restrict__ sshift, float* __restrict__ out) {
  __shared__ __align__(16) float wlds[C_OUT * K_PAD];
  __shared__ __align__(16) float flds[CHUNK_PTS * K_PAD];
  __shared__ float accw[8][C_OUT];  // per-wave 16-point channel sums

  const int t = threadIdx.x;
  const int wave = t >> 5;
  const int lane = t & 31;
  const int m = lane & 15;
  const int khalf = lane >> 4;

  stage_weights(wlds, W, b, t);
  __syncthreads();

  v2f breg[4][3];
  load_breg(wlds, breg, m, khalf);

  float scl[4], shf[4];
#pragma unroll
  for (int tl = 0; tl < 4; ++tl) {
    int n = tl * 16 + m;
    scl[tl] = sshift[n];
    shf[tl] = sshift[C_OUT + n];
  }

  for (int c = blockIdx.x; c < N_CHUNKS; c += gridDim.x) {
    const int g0 = c * CHUNK_PTS;
    stage_features(flds, vf, vnp, g0, t);
    __syncthreads();

    v8f z = {};
    v8f acc[4] = {z, z, z, z};
    const float* arow = &flds[(wave * 16 + m) * K_PAD];
#pragma unroll
    for (int ks = 0; ks < 3; ++ks) {
      v2f a;
      a.x = arow[ks * 4 + 2 * khalf];
      a.y = arow[ks * 4 + 2 * khalf + 1];
#pragma unroll
      for (int tl = 0; tl < 4; ++tl) acc[tl] = wmma_k4(a, breg[tl][ks], acc[tl]);
    }

    // This wave's 16 points all lie in one voxel (16 | 32).
    const int wbase = g0 + wave * 16;
    const int vx = wbase >> 5;
    int np = vnp[vx];
    np = (np < P_NUM) ? np : P_NUM;
    const int pbase = (wbase & 31) + 8 * khalf;  // point offset of VGPR r: pbase + r

    float lsum[4];
#pragma unroll
    for (int tl = 0; tl < 4; ++tl) {
      float ls = 0.f;
#pragma unroll
      for (int r = 0; r < 8; ++r) {
        float y = fmaxf(acc[tl][r] * scl[tl] + shf[tl], 0.0f);
        ls += ((pbase + r) < np) ? y : 0.0f;
      }
      ls += __shfl_xor(ls, 16);  // fold M halves -> 16-point sum per channel
      lsum[tl] = ls;
    }
    if (lane < 16) {
#pragma unroll
      for (int tl = 0; tl < 4; ++tl) accw[wave][tl * 16 + lane] = lsum[tl];
    }
    __syncthreads();

    // 4 voxels x 64 channels = 256 outputs: one per thread, deterministic order.
    {
      const int vl = t >> 6;
      const int n = t & 63;
      const int gv = (g0 >> 5) + vl;
      int npv = vnp[gv];
      npv = (npv < P_NUM) ? npv : P_NUM;
      float tot = accw[2 * vl][n] + accw[2 * vl + 1][n];
      out[gv * C_OUT + n] = (npv > 0) ? tot / (float)npv : 0.0f;
    }
    __syncthreads();  // before next chunk overwrites flds/accw
  }
}

extern "C" void kernel_launch(void* const* d_in, const int* in_sizes, int n_in,
                              void* d_out, int out_size, void* d_ws, size_t ws_size,
                              hipStream_t stream) {
  const float* vf    = (const float*)d_in[0];  // (V, P, C_IN)
  const int*   vnp   = (const int*)d_in[1];    // (V,)
  const float* W     = (const float*)d_in[2];  // (C_OUT, C_IN)
  const float* b     = (const float*)d_in[3];  // (C_OUT,)
  const float* gamma = (const float*)d_in[4];  // (C_OUT,)
  const float* beta  = (const float*)d_in[5];  // (C_OUT,)
  float* out = (float*)d_out;                  // (V, C_OUT)

  float* ws = (float*)d_ws;
  float* psum   = ws;                          // [GRID1][C_OUT]
  float* psq    = ws + GRID1 * C_OUT;          // [GRID1][C_OUT]
  float* sshift = ws + 2 * GRID1 * C_OUT;      // scale[64] then shift[64]

  vfe_stats_kernel<<<GRID1, NTHREADS, 0, stream>>>(vf, vnp, W, b, psum, psq);
  vfe_finalize_kernel<<<C_OUT, NTHREADS, 0, stream>>>(vnp, gamma, beta, psum, psq,
                                                      sshift);
  vfe_output_kernel<<<GRID1, NTHREADS, 0, stream>>>(vf, vnp, W, b, sshift, out);
}